// QuantumLayer_21182778704874
// MI455X (gfx1250) — compile-verified
//
#include <hip/hip_runtime.h>
#include <hip/hip_bf16.h>

typedef __attribute__((ext_vector_type(2))) float v2f;
typedef __attribute__((ext_vector_type(8))) float v8f;

// ---------------------------------------------------------------------------
// Prep kernel: build the fixed 16x16 complex unitary U for the 2 entangling
// layers (RY,RZ per wire + CNOT ring) by simulating the 16 basis columns,
// then store its real 32x32 embedding W in the WMMA-B operand layout:
//   Bmat[((kstep*2 + nhalf)*32 + lane)*2 + v]
// where for lane<16:  K = kstep*4 + v,     N = nhalf*16 + lane
//       for lane>=16: K = kstep*4 + 2 + v, N = nhalf*16 + (lane-16)
// W(j,n): j<16,n<16: Ur[n][j]; j>=16,n<16: -Ui[n][j-16];
//         j<16,n>=16: Ui[n-16][j]; j>=16,n>=16: Ur[n-16][j-16]
// Qubit q lives on bit (3-q) of the basis index (mask = 8>>q).
// Runs on ONE thread once; unrolling suppressed to keep static code small.
// ---------------------------------------------------------------------------
__global__ void qlayer_prep(const float* __restrict__ angles,
                            float* __restrict__ Bmat) {
  if (threadIdx.x != 0 || blockIdx.x != 0) return;

  float ur[16][16], ui[16][16];

#pragma unroll 1
  for (int col = 0; col < 16; ++col) {
    float re[16], im[16];
#pragma unroll 1
    for (int i = 0; i < 16; ++i) { re[i] = 0.f; im[i] = 0.f; }
    re[col] = 1.f;

#pragma unroll 1
    for (int layer = 0; layer < 2; ++layer) {
#pragma unroll 1
      for (int q = 0; q < 4; ++q) {
        const int mask = 8 >> q;
        // RY(theta): [[cy,-sy],[sy,cy]]
        const float ty = angles[(layer * 4 + q) * 2 + 0] * 0.5f;
        const float cy = cosf(ty), sy = sinf(ty);
#pragma unroll 1
        for (int b = 0; b < 16; ++b) {
          if (b & mask) continue;
          const int b1 = b | mask;
          const float r0 = re[b], i0 = im[b], r1 = re[b1], i1 = im[b1];
          re[b]  = cy * r0 - sy * r1;  im[b]  = cy * i0 - sy * i1;
          re[b1] = sy * r0 + cy * r1;  im[b1] = sy * i0 + cy * i1;
        }
        // RZ(theta): diag(e^{-i tz}, e^{+i tz})
        const float tz = angles[(layer * 4 + q) * 2 + 1] * 0.5f;
        const float cz = cosf(tz), sz = sinf(tz);
#pragma unroll 1
        for (int b = 0; b < 16; ++b) {
          const float r = re[b], i = im[b];
          if (b & mask) { re[b] = r * cz - i * sz; im[b] = r * sz + i * cz; }
          else          { re[b] = r * cz + i * sz; im[b] = i * cz - r * sz; }
        }
      }
      // CNOT chain (0,1),(1,2),(2,3) + ring (3,0)
      const int ctrl[4] = {0, 1, 2, 3};
      const int tgt[4]  = {1, 2, 3, 0};
#pragma unroll 1
      for (int g = 0; g < 4; ++g) {
        const int cm = 8 >> ctrl[g], tm = 8 >> tgt[g];
#pragma unroll 1
        for (int b = 0; b < 16; ++b) {
          if ((b & cm) && !(b & tm)) {
            const int b2 = b | tm;
            float t;
            t = re[b]; re[b] = re[b2]; re[b2] = t;
            t = im[b]; im[b] = im[b2]; im[b2] = t;
          }
        }
      }
    }
#pragma unroll 1
    for (int n = 0; n < 16; ++n) { ur[n][col] = re[n]; ui[n][col] = im[n]; }
  }

  // Swizzle into WMMA B-operand layout
#pragma unroll 1
  for (int ks = 0; ks < 8; ++ks)
#pragma unroll 1
    for (int nh = 0; nh < 2; ++nh)
#pragma unroll 1
      for (int lane = 0; lane < 32; ++lane)
#pragma unroll 1
        for (int v = 0; v < 2; ++v) {
          const int k = ks * 4 + ((lane >= 16) ? 2 : 0) + v;
          const int n = nh * 16 + (lane & 15);
          const int jr = k & 15, nr = n & 15;
          float val;
          if (k < 16) val = (n < 16) ?  ur[nr][jr] : ui[nr][jr];
          else        val = (n < 16) ? -ui[nr][jr] : ur[nr][jr];
          Bmat[((ks * 2 + nh) * 32 + lane) * 2 + v] = val;
        }
}

// ---------------------------------------------------------------------------
// Compile-time product-state component n of the RX-embedded input state
// (real 32-vector: n<16 -> Re(amp_n), n>=16 -> Im(amp_{n-16})).
//   amp_b = (-i)^popcount(b) * P_b,  P_b = h[b>>2] * l[b&3]
// where h = {c0c1, c0s1, s0c1, s0s1}, l = {c2c3, c2s3, s2c3, s2s3}.
// Sign / structural zero fully folded at compile time.
// ---------------------------------------------------------------------------
template <int n>
__device__ __forceinline__ float psi_c(const float* h, const float* l) {
  constexpr int b = n & 15;
  constexpr int w = ((b & 1) + ((b >> 1) & 1) + ((b >> 2) & 1) + ((b >> 3) & 1)) & 3;
  constexpr int sgn = (n < 16) ? ((w == 0) ? 1 : ((w == 2) ? -1 : 0))
                               : ((w == 1) ? -1 : ((w == 3) ? 1 : 0));
  if constexpr (sgn == 0) return 0.0f;
  else if constexpr (sgn == 1) return h[b >> 2] * l[b & 3];
  else return -(h[b >> 2] * l[b & 3]);
}

// ds_swizzle XOR butterfly (group-of-32 mode: and=0x1f, or=0, xor=j).
template <int imm>
__device__ __forceinline__ float swz(float v) {
  return __int_as_float(__builtin_amdgcn_ds_swizzle(__float_as_int(v), imm));
}

// ---------------------------------------------------------------------------
// Main kernel: one wave handles 16 batch rows via V_WMMA_F32_16X16X4_F32.
//   acc0 = real parts of final amplitudes (N = 0..15 across lanes)
//   acc1 = imag parts                      (N = 16..31)
// D layout: VGPR r holds M=r (lanes 0-15) and M=r+8 (lanes 16-31).
// After p = re^2+im^2, a 4-stage ds_swizzle Walsh-Hadamard over the 16-lane
// halves yields X[t] = sum_b (-1)^{popcount(t&b)} p_b in lane t;
// <Z_q> = X[8>>q], stored by lanes 8,4,2,1 (and 24,20,18,17).
// ---------------------------------------------------------------------------
__global__ void __launch_bounds__(256)
qlayer_wmma(const float* __restrict__ inputs, const float* __restrict__ Bmat,
            float* __restrict__ out, int batch) {
  const int lane = threadIdx.x & 31;
  const int wave = threadIdx.x >> 5;
  const long tile = (long)blockIdx.x * (blockDim.x >> 5) + wave;
  const long mbase = tile * 16;
  if (mbase >= batch) return;

  const int l = lane & 15;
  const int hi = lane >> 4;  // K-half selector for the A operand
  long m = mbase + l;
  if (m >= batch) m = batch - 1;  // safe duplicate load for ragged tail

  const float4 x = reinterpret_cast<const float4*>(inputs)[m];
  const float c0 = __cosf(x.x * 0.5f), s0 = __sinf(x.x * 0.5f);
  const float c1 = __cosf(x.y * 0.5f), s1 = __sinf(x.y * 0.5f);
  const float c2 = __cosf(x.z * 0.5f), s2 = __sinf(x.z * 0.5f);
  const float c3 = __cosf(x.w * 0.5f), s3 = __sinf(x.w * 0.5f);

  // Pair-product tables (8 muls); P_b = h[b>>2]*l[b&3].
  const float h[4] = {c0 * c1, c0 * s1, s0 * c1, s0 * s1};
  const float q[4] = {c2 * c3, c2 * s3, s2 * c3, s2 * s3};

  // A operands: lane half hi needs K = 4*ks + 2*hi + {0,1}; all indices are
  // compile-time per branch, so signs/zeros fold away (8 nonzero comps/lane).
  // Branch reconverges before the WMMAs (EXEC all-ones at matrix ops).
  v2f aop[8];
  if (hi == 0) {
    aop[0].x = psi_c<0>(h, q);   aop[0].y = psi_c<1>(h, q);
    aop[1].x = psi_c<4>(h, q);   aop[1].y = psi_c<5>(h, q);
    aop[2].x = psi_c<8>(h, q);   aop[2].y = psi_c<9>(h, q);
    aop[3].x = psi_c<12>(h, q);  aop[3].y = psi_c<13>(h, q);
    aop[4].x = psi_c<16>(h, q);  aop[4].y = psi_c<17>(h, q);
    aop[5].x = psi_c<20>(h, q);  aop[5].y = psi_c<21>(h, q);
    aop[6].x = psi_c<24>(h, q);  aop[6].y = psi_c<25>(h, q);
    aop[7].x = psi_c<28>(h, q);  aop[7].y = psi_c<29>(h, q);
  } else {
    aop[0].x = psi_c<2>(h, q);   aop[0].y = psi_c<3>(h, q);
    aop[1].x = psi_c<6>(h, q);   aop[1].y = psi_c<7>(h, q);
    aop[2].x = psi_c<10>(h, q);  aop[2].y = psi_c<11>(h, q);
    aop[3].x = psi_c<14>(h, q);  aop[3].y = psi_c<15>(h, q);
    aop[4].x = psi_c<18>(h, q);  aop[4].y = psi_c<19>(h, q);
    aop[5].x = psi_c<22>(h, q);  aop[5].y = psi_c<23>(h, q);
    aop[6].x = psi_c<26>(h, q);  aop[6].y = psi_c<27>(h, q);
    aop[7].x = psi_c<30>(h, q);  aop[7].y = psi_c<31>(h, q);
  }

  v8f acc0 = {0.f, 0.f, 0.f, 0.f, 0.f, 0.f, 0.f, 0.f};
  v8f acc1 = {0.f, 0.f, 0.f, 0.f, 0.f, 0.f, 0.f, 0.f};

  const v2f* __restrict__ Bv = reinterpret_cast<const v2f*>(Bmat);

#pragma unroll
  for (int ks = 0; ks < 8; ++ks) {
    const v2f b0 = Bv[(ks * 2 + 0) * 32 + lane];
    const v2f b1 = Bv[(ks * 2 + 1) * 32 + lane];
    acc0 = __builtin_amdgcn_wmma_f32_16x16x4_f32(false, aop[ks], false, b0,
                                                 (short)0, acc0, false, false);
    acc1 = __builtin_amdgcn_wmma_f32_16x16x4_f32(false, aop[ks], false, b1,
                                                 (short)0, acc1, false, false);
  }

  // Per-stage butterfly signs (hoisted): p' = sgn_j * p + swizzle_xor_j(p)
  // gives p+partner on bit-clear lanes and partner-p on bit-set lanes.
  const float sg1 = (lane & 1) ? -1.f : 1.f;
  const float sg2 = (lane & 2) ? -1.f : 1.f;
  const float sg4 = (lane & 4) ? -1.f : 1.f;
  const float sg8 = (lane & 8) ? -1.f : 1.f;

  // Writer lanes: l in {8,4,2,1} -> qubit q = 3 - log2(l).
  const bool writer = (l != 0) && ((l & (l - 1)) == 0);
  const int qidx = writer ? (3 - (31 - __clz((unsigned)l))) : 0;

  // Per-row probability + signed lane reduction + store of <Z_q>.
#pragma unroll
  for (int r = 0; r < 8; ++r) {
    float p = acc0[r] * acc0[r] + acc1[r] * acc1[r];
    float t;
    t = swz<0x041f>(p); p = fmaf(sg1, p, t);  // SWAPX1
    t = swz<0x081f>(p); p = fmaf(sg2, p, t);  // SWAPX2
    t = swz<0x101f>(p); p = fmaf(sg4, p, t);  // SWAPX4
    t = swz<0x201f>(p); p = fmaf(sg8, p, t);  // SWAPX8
    const long mrow = mbase + r + hi * 8;  // D layout: M = r (+8 upper lanes)
    if (writer && mrow < batch) out[mrow * 4 + qidx] = p;
  }
}

extern "C" void kernel_launch(void* const* d_in, const int* in_sizes, int n_in,
                              void* d_out, int out_size, void* d_ws,
                              size_t ws_size, hipStream_t stream) {
  const float* inputs = (const float*)d_in[0];  // [B,4]
  const float* angles = (const float*)d_in[1];  // [2,4,2]
  float* out = (float*)d_out;                   // [B,4]
  float* Bmat = (float*)d_ws;                   // 1024 floats (4 KB)

  const int batch = in_sizes[0] / 4;

  qlayer_prep<<<1, 32, 0, stream>>>(angles, Bmat);

  const int threads = 256;                 // 8 waves -> 128 rows per block
  const int rows_per_block = (threads / 32) * 16;
  const int blocks = (batch + rows_per_block - 1) / rows_per_block;
  qlayer_wmma<<<blocks, threads, 0, stream>>>(inputs, Bmat, out, batch);
}